// QueryAndGroup_69148973465940
// MI455X (gfx1250) — compile-verified
//
#include <hip/hip_runtime.h>
#include <hip/hip_bf16.h>

typedef __attribute__((ext_vector_type(2))) float v2f;
typedef __attribute__((ext_vector_type(8))) float v8f;

// Problem constants (from the reference)
static constexpr int   BATCH  = 2;
static constexpr int   NPTS   = 16384;
static constexpr int   NCTR   = 2048;
static constexpr int   NCH    = 64;
static constexpr int   KNN    = 32;
static constexpr float RADIUS = 0.2f;

// ---------------------------------------------------------------------------
// Kernel 1: ball query. One wave (32 lanes) owns 16 centers.
// Per iteration: 32 unique points loaded coalesced (one b96 per lane),
// 4 shfl_xor(16) redistribute coords across half-waves, two
// V_WMMA_F32_16X16X4_F32 produce two 16x16 tiles of (|p|^2 - 2 c.p),
// ballot-extract first-K in-order indices, uniform early exit.
// Loads are software-pipelined one iteration ahead.
// ---------------------------------------------------------------------------
__global__ __launch_bounds__(32)
void QueryAndGroup_ballquery_kernel(const float* __restrict__ pts,
                                    const float* __restrict__ ctr,
                                    int* __restrict__ idx_ws)
{
    const int lane  = threadIdx.x;            // 0..31 (wave32)
    const int wid   = blockIdx.x;             // 0..255
    const int b     = wid >> 7;               // /128 groups per batch
    const int gbase = (wid & 127) << 4;       // 16 centers per wave
    const int m     = lane & 15;              // center slot in both halves
    const bool lo   = (lane < 16);

    // Load center m (both lane halves load the same center).
    const float* cp = ctr + ((size_t)b * NCTR + gbase + m) * 3;
    const float cx = cp[0], cy = cp[1], cz = cp[2];
    const float thr = RADIUS * RADIUS - (cx * cx + cy * cy + cz * cz);

    // A matrix 16x4 f32: VGPR0 = {K=0 | K=2}, VGPR1 = {K=1 | K=3} across halves.
    v2f a;
    a.x = lo ? cx : cz;
    a.y = lo ? cy : 1.0f;

    // Per-row thresholds aligned with C layout: VGPR v -> rows v | v+8.
    float tv[8];
#pragma unroll
    for (int v = 0; v < 8; ++v)
        tv[v] = __shfl(thr, lo ? v : (v + 8));

    const float* pbase = pts + (size_t)b * NPTS * 3;
    int* wsp = idx_ws + ((size_t)b * NCTR + gbase + m) * KNN;

    int cnt = 0;
    int first = -1;
    const v8f czero = {};
    constexpr int NT = NPTS / 32;             // 512 iterations max

    // Pipeline prologue: load points 0..31 (lane -> point lane), coalesced.
    float px, py, pz;
    {
        const float* pp = pbase + (size_t)lane * 3;
        px = pp[0]; py = pp[1]; pz = pp[2];
    }

    for (int tile = 0; tile < NT; ++tile) {
        const int j = tile * 32;

        // Prefetch next iteration's 32 points (clamped; extra load harmless).
        const int jn = (tile + 1 < NT) ? (j + 32) : j;
        const float* pn = pbase + (size_t)(jn + lane) * 3;
        const float nx = pn[0], ny = pn[1], nz = pn[2];

        const float psq = px * px + py * py + pz * pz;

        // Cross-half exchange: lane i gets lane i^16's values.
        const float xo = __shfl_xor(px, 16);
        const float yo = __shfl_xor(py, 16);
        const float zo = __shfl_xor(pz, 16);
        const float qo = __shfl_xor(psq, 16);

        // B tiles (4x16 f32, mirrors A layout). Tile0 = points j..j+15,
        // Tile1 = points j+16..j+31.
        v2f b0, b1;
        b0.x = lo ? (-2.0f * px) : (-2.0f * zo);
        b0.y = lo ? (-2.0f * py) : qo;
        b1.x = lo ? (-2.0f * xo) : (-2.0f * pz);
        b1.y = lo ? (-2.0f * yo) : psq;

        // D[M][N] = |p_N|^2 - 2 * c_M . p_N
        v8f d0 = __builtin_amdgcn_wmma_f32_16x16x4_f32(
            false, a, false, b0, (short)0, czero, false, false);
        v8f d1 = __builtin_amdgcn_wmma_f32_16x16x4_f32(
            false, a, false, b1, (short)0, czero, false, false);

        // Per-VGPR ballots: bits 0-15 = center v, bits 16-31 = center v+8.
        unsigned pack0 = 0u, pack1 = 0u;
#pragma unroll
        for (int v = 0; v < 8; ++v) {
            unsigned bal0 = (unsigned)__ballot(d0[v] < tv[v]);
            unsigned bal1 = (unsigned)__ballot(d1[v] < tv[v]);
            if (lane == v) { pack0 = bal0; pack1 = bal1; }
        }

        // Route each center's 16-bit masks to lane m; concat to 32-bit mask
        // over points j..j+31 (bit k -> point j+k).
        const int sh = (lane & 8) ? 16 : 0;
        unsigned mm0 = ((unsigned)__shfl((int)pack0, lane & 7) >> sh) & 0xFFFFu;
        unsigned mm1 = ((unsigned)__shfl((int)pack1, lane & 7) >> sh) & 0xFFFFu;
        unsigned mm = mm0 | (mm1 << 16);

        if (lane < 16) {
            while (mm && cnt < KNN) {
                const int bpos = __ffs(mm) - 1;
                const int pidx = j + bpos;
                if (cnt == 0) first = pidx;
                wsp[cnt++] = pidx;
                mm &= mm - 1u;
            }
        }

        if (__all((lane >= 16) | (cnt >= KNN))) break;  // uniform early exit

        px = nx; py = ny; pz = nz;
    }

    // Pad remaining slots with first hit (or 0 if none).
    if (lane < 16) {
        const int pad = (first < 0) ? 0 : first;
        for (int k = cnt; k < KNN; ++k) wsp[k] = pad;
    }
}

// ---------------------------------------------------------------------------
// Kernel 2: gather. One wave per (batch, center); lane = k slot.
// Writes grouped_xyz (2,3,2048,32) then grouped_features (2,64,2048,32),
// concatenated flat in d_out. Row stores are 128B contiguous; channel loop
// unrolled so many independent gather loads stay in flight.
// ---------------------------------------------------------------------------
__global__ __launch_bounds__(256)
void QueryAndGroup_gather_kernel(const float* __restrict__ pts,
                                 const float* __restrict__ ctr,
                                 const float* __restrict__ feat,
                                 const int* __restrict__ idx_ws,
                                 float* __restrict__ out)
{
    const int lane = threadIdx.x & 31;            // k
    const int w    = threadIdx.x >> 5;
    const int pair = blockIdx.x * 8 + w;          // 0..BATCH*NCTR-1
    const int b    = pair >> 11;                  // / NCTR
    const int g    = pair & (NCTR - 1);

    const int id = idx_ws[((size_t)b * NCTR + g) * KNN + lane];

    // grouped_xyz = (p[idx] - c) / RADIUS
    const float* p = pts + ((size_t)b * NPTS + id) * 3;
    const float* c = ctr + ((size_t)b * NCTR + g) * 3;
    const float inv = 1.0f / RADIUS;
#pragma unroll
    for (int dd = 0; dd < 3; ++dd)
        out[(((size_t)b * 3 + dd) * NCTR + g) * KNN + lane] = (p[dd] - c[dd]) * inv;

    // grouped_features = features[:, :, idx]
    float* fout = out + (size_t)BATCH * 3 * NCTR * KNN +
                  (((size_t)b * NCH) * NCTR + g) * KNN + lane;
    const float* frow = feat + (size_t)b * NCH * NPTS + id;
#pragma unroll 8
    for (int ch = 0; ch < NCH; ++ch)
        fout[(size_t)ch * NCTR * KNN] = frow[(size_t)ch * NPTS];
}

extern "C" void kernel_launch(void* const* d_in, const int* in_sizes, int n_in,
                              void* d_out, int out_size, void* d_ws, size_t ws_size,
                              hipStream_t stream) {
    const float* points  = (const float*)d_in[0];   // (2,16384,3)
    const float* centers = (const float*)d_in[1];   // (2,2048,3)
    const float* feats   = (const float*)d_in[2];   // (2,64,16384)
    float* out = (float*)d_out;
    int* idx_ws = (int*)d_ws;                       // 2*2048*32 ints = 512 KB

    // 256 waves total: each handles 16 centers of one batch.
    QueryAndGroup_ballquery_kernel<<<(BATCH * NCTR) / 16, 32, 0, stream>>>(
        points, centers, idx_ws);

    // 4096 (b,g) pairs, 8 waves per block.
    QueryAndGroup_gather_kernel<<<(BATCH * NCTR) / 8, 256, 0, stream>>>(
        points, centers, feats, idx_ws, out);
}